// Hydra_56633438765296
// MI455X (gfx1250) — compile-verified
//
#include <hip/hip_runtime.h>
#include <math.h>

// ---------------- problem constants ----------------
#define B_TOTAL 32768
#define NJETS   10
#define NCOMBOS 22
#define NSEL    6
#define FF      200
#define FF_PAD  208          // 13 tiles of 16 (W1/b1 zero-padded to this)
#define N_ISR   32           // NJETS + NCOMBOS
#define N_DEC   28           // NSEL + NCOMBOS
#define F_ISR   160          // N_ISR * 5
#define F_DEC   140          // N_DEC * 5
#define SA_ISR  164          // LDS row stride (floats), gcd(164,64)=4 -> conflict free
#define SA_DEC  148          // gcd(148,64)=4 -> conflict free
#define BN_EPS  1e-5f
#define STAT_BLOCKS 256

typedef __attribute__((ext_vector_type(2))) float v2f;
typedef __attribute__((ext_vector_type(8))) float v8f;

// =====================================================================
// Fused CoLa + LoLa feature builder. One wave32 per event; lane i owns
// particle i (n<=32). Pairwise Minkowski gram via lane broadcasts —
// never materializes [b,n,n].
// =====================================================================
__global__ __launch_bounds__(256) void feat_kernel(
    const float* __restrict__ vecs, int vstride,      // floats per event row
    int n0, int n, int F,
    const float* __restrict__ cola_w,                 // [NCOMBOS, n0]
    const float* __restrict__ we_w,                   // [n, n]
    const float* __restrict__ wd_w,                   // [n, n]
    float* __restrict__ feat)                         // [B, F]
{
    const int wave = threadIdx.x >> 5;
    const int lane = threadIdx.x & 31;
    const int b = blockIdx.x * (blockDim.x >> 5) + wave;

    const float* vrow = vecs + (size_t)b * vstride;
    float e = 0.f, px = 0.f, py = 0.f, pz = 0.f;
    if (lane < n0) {
        float4 v = *(const float4*)(vrow + lane * 4);
        e = v.x; px = v.y; py = v.z; pz = v.w;
    }

    // CoLa: lanes [n0, n) build learned combos from broadcast 4-vectors
    float ce = e, cpx = px, cpy = py, cpz = pz;
    if (lane >= n0) { ce = cpx = cpy = cpz = 0.f; }
    for (int j = 0; j < n0; ++j) {
        float ej  = __shfl(e,  j);
        float pxj = __shfl(px, j);
        float pyj = __shfl(py, j);
        float pzj = __shfl(pz, j);
        if (lane >= n0 && lane < n) {
            float w = cola_w[(lane - n0) * n0 + j];
            ce += w * ej; cpx += w * pxj; cpy += w * pyj; cpz += w * pzj;
        }
    }
    e = ce; px = cpx; py = cpy; pz = cpz;

    float m2 = e * e - px * px - py * py - pz * pz;
    float pt = sqrtf(px * px + py * py);

    // LoLa: weighted E and weighted pairwise Minkowski distance
    float we_acc = 0.f, wd_acc = 0.f;
    for (int j = 0; j < n; ++j) {
        float ej  = __shfl(e,  j);
        float pxj = __shfl(px, j);
        float pyj = __shfl(py, j);
        float pzj = __shfl(pz, j);
        float m2j = __shfl(m2, j);
        if (lane < n) {
            float gram = e * ej - px * pxj - py * pyj - pz * pzj;
            float d2   = m2 + m2j - 2.f * gram;
            we_acc = fmaf(we_w[lane * n + j], ej, we_acc);
            wd_acc = fmaf(wd_w[lane * n + j], d2, wd_acc);
        }
    }
    if (lane < n) {
        float* o = feat + (size_t)b * F + lane * 5;
        o[0] = m2; o[1] = pt; o[2] = e; o[3] = we_acc; o[4] = wd_acc;
    }
}

// =====================================================================
// Zero-pad W1 [FF,F] -> [FF_PAD,F] and b1 [FF] -> [FF_PAD] so the WMMA
// inner loop needs no bounds guard (no EXEC juggling around loads).
// =====================================================================
__global__ __launch_bounds__(256) void pad_w1_kernel(
    const float* __restrict__ w1, const float* __restrict__ b1, int F,
    float* __restrict__ w1p, float* __restrict__ b1p)
{
    int i = blockIdx.x * 256 + threadIdx.x;
    int total = FF_PAD * F;
    if (i < total) {
        int n = i / F;                 // w1p[n*F+k]; source identical layout for n<FF
        w1p[i] = (n < FF) ? w1[i] : 0.f;
    }
    if (i < FF_PAD) b1p[i] = (i < FF) ? b1[i] : 0.f;
}

// =====================================================================
// Per-block partial sums / sumsq for batch-wide BatchNorm.
// Deterministic (no float atomics) so graph replays are bit-stable.
// =====================================================================
__global__ __launch_bounds__(256) void stats_kernel(
    const float* __restrict__ feat, int F, int rows_per_blk,
    float* __restrict__ partials)                     // [STAT_BLOCKS, 2, F]
{
    const int t = threadIdx.x;
    if (t >= F) return;
    const int r0 = blockIdx.x * rows_per_blk;
    float s = 0.f, s2 = 0.f;
    for (int r = 0; r < rows_per_blk; ++r) {
        float x = feat[(size_t)(r0 + r) * F + t];
        s += x;
        s2 = fmaf(x, x, s2);
    }
    partials[(size_t)blockIdx.x * 2 * F + t]     = s;
    partials[(size_t)blockIdx.x * 2 * F + F + t] = s2;
}

// Fold BN into per-feature scale/shift: y = x*scale + shift
__global__ __launch_bounds__(256) void finalize_kernel(
    const float* __restrict__ partials, int F,
    const float* __restrict__ gamma, const float* __restrict__ beta,
    float* __restrict__ scale, float* __restrict__ shift)
{
    const int t = threadIdx.x;
    if (t >= F) return;
    float s = 0.f, s2 = 0.f;
    for (int b = 0; b < STAT_BLOCKS; ++b) {
        s  += partials[(size_t)b * 2 * F + t];
        s2 += partials[(size_t)b * 2 * F + F + t];
    }
    const float inv_b = 1.f / (float)B_TOTAL;
    float mean = s * inv_b;
    float var  = s2 * inv_b - mean * mean;
    float sc   = gamma[t] * rsqrtf(var + BN_EPS);
    scale[t] = sc;
    shift[t] = beta[t] - mean * sc;
}

// =====================================================================
// BN-normalize -> FC1 via V_WMMA_F32_16X16X4_F32 -> ReLU -> FC2 ->
// output (+ optional top-6 select/gather for the ISR head).
// Block = 128 threads (4 waves) x 16 batch rows. Each wave owns N-tiles
// {w, w+4, w+8, (12)} of the 13 FF tiles, chaining fp32 WMMA over the
// fully-unrolled K/4 steps with bias preloaded into the accumulator.
// W1 is pre-padded, so the inner loop is pure ds_load/global_load/wmma.
// =====================================================================
template<int F, int SA, int NOUT, bool DO_SEL>
__global__ __launch_bounds__(128) void fc_kernel(
    const float* __restrict__ feat,
    const float* __restrict__ scale, const float* __restrict__ shift,
    const float* __restrict__ w1p, const float* __restrict__ b1p, // [FF_PAD,F],[FF_PAD]
    const float* __restrict__ w2, const float* __restrict__ b2,   // [NOUT,FF],[NOUT]
    int out_off,
    float* __restrict__ out,                                      // [B,21]
    const float* __restrict__ vecs,                               // [B,10,4] (sel only)
    float* __restrict__ sel)                                      // [B,6,4]  (sel only)
{
    __shared__ float ldsA[16 * SA];         // normalized A tile
    __shared__ float ldsH[16 * FF_PAD];     // hidden layer tile
    __shared__ float ldsS[16 * 16];         // scores for top-k

    const int b0  = blockIdx.x * 16;
    const int tid = threadIdx.x;

    // stage + normalize 16 rows of features into LDS (BN applied once here)
    #pragma unroll
    for (int i = tid; i < 16 * F; i += 128) {
        int r = i / F, k = i - r * F;
        float x = feat[(size_t)(b0 + r) * F + k];
        ldsA[r * SA + k] = fmaf(x, scale[k], shift[k]);
    }
    __syncthreads();

    const int wave = tid >> 5;
    const int lane = tid & 31;
    const int l15  = lane & 15;
    const int hlf  = lane >> 4;
    const int NT   = FF_PAD / 16;           // 13

    for (int nt = wave; nt < NT; nt += 4) {
        const int nn = nt * 16 + l15;       // output unit for this lane (B/C col)
        float bias = b1p[nn];
        v8f acc;
        #pragma unroll
        for (int i = 0; i < 8; ++i) acc[i] = bias;

        const float* ldsArow = ldsA + l15 * SA + 2 * hlf;  // A: M = lane&15
        const float* w1row   = w1p + (size_t)nn * F + 2 * hlf;
        #pragma unroll
        for (int k0 = 0; k0 < F; k0 += 4) {
            v2f a  = *(const v2f*)(ldsArow + k0);  // {A[m][kb], A[m][kb+1]}
            v2f bf = *(const v2f*)(w1row + k0);    // {B[kb][n], B[kb+1][n]}
            acc = __builtin_amdgcn_wmma_f32_16x16x4_f32(
                      false, a, false, bf, (short)0, acc, false, false);
        }
        // ReLU, scatter 16x16 D tile into LDS (C/D: vgpr i -> M=i+8*half)
        #pragma unroll
        for (int i = 0; i < 8; ++i) {
            int m = i + 8 * hlf;
            float h = acc[i] > 0.f ? acc[i] : 0.f;
            ldsH[m * FF_PAD + nt * 16 + l15] = h;
        }
    }
    __syncthreads();

    // FC2: 16*NOUT small dot products out of LDS (broadcast-friendly reads)
    for (int t = tid; t < 16 * NOUT; t += 128) {
        int m = t & 15, o = t >> 4;
        float acc2 = b2[o];
        const float* hrow = ldsH + m * FF_PAD;
        const float* wrow = w2 + (size_t)o * FF;
        #pragma unroll 8
        for (int k = 0; k < FF; ++k) acc2 = fmaf(hrow[k], wrow[k], acc2);
        out[(size_t)(b0 + m) * 21 + out_off + o] = acc2;
        ldsS[m * 16 + o] = acc2;
    }

    // ISR head only: top-6 (ties -> lowest index, like lax.top_k), indices
    // sorted ascending (np.delete order), gather selected 4-vectors.
    if (DO_SEL) {
        __syncthreads();
        if (tid < 16) {
            const int m = tid;
            float sc_[NJETS];
            #pragma unroll
            for (int j = 0; j < NJETS; ++j) sc_[j] = ldsS[m * 16 + j];
            int idx[NSEL];
            #pragma unroll
            for (int s = 0; s < NSEL; ++s) {
                int best = 0; float bv = sc_[0];
                #pragma unroll
                for (int j = 1; j < NJETS; ++j)
                    if (sc_[j] > bv) { bv = sc_[j]; best = j; }
                idx[s] = best;
                sc_[best] = -3.0e38f;
            }
            #pragma unroll
            for (int a = 1; a < NSEL; ++a) {        // insertion sort ascending
                int key = idx[a], c = a - 1;
                while (c >= 0 && idx[c] > key) { idx[c + 1] = idx[c]; --c; }
                idx[c + 1] = key;
            }
            const float* vrow = vecs + (size_t)(b0 + m) * NJETS * 4;
            float*       srow = sel  + (size_t)(b0 + m) * NSEL * 4;
            #pragma unroll
            for (int s = 0; s < NSEL; ++s)
                *(float4*)(srow + s * 4) = *(const float4*)(vrow + idx[s] * 4);
        }
    }
}

// =====================================================================
extern "C" void kernel_launch(void* const* d_in, const int* in_sizes, int n_in,
                              void* d_out, int out_size, void* d_ws, size_t ws_size,
                              hipStream_t stream)
{
    (void)in_sizes; (void)n_in; (void)out_size; (void)ws_size;
    const float* vectors    = (const float*)d_in[0];
    const float* isr_cola_w = (const float*)d_in[1];
    const float* isr_we     = (const float*)d_in[2];
    const float* isr_wd     = (const float*)d_in[3];
    const float* isr_bn_g   = (const float*)d_in[4];
    const float* isr_bn_b   = (const float*)d_in[5];
    const float* isr_w1     = (const float*)d_in[6];
    const float* isr_b1     = (const float*)d_in[7];
    const float* isr_w2     = (const float*)d_in[8];
    const float* isr_b2     = (const float*)d_in[9];
    const float* dec_cola_w = (const float*)d_in[10];
    const float* dec_we     = (const float*)d_in[11];
    const float* dec_wd     = (const float*)d_in[12];
    const float* dec_bn_g   = (const float*)d_in[13];
    const float* dec_bn_b   = (const float*)d_in[14];
    const float* dec_w1     = (const float*)d_in[15];
    const float* dec_b1     = (const float*)d_in[16];
    const float* dec_w2     = (const float*)d_in[17];
    const float* dec_b2     = (const float*)d_in[18];
    float* out = (float*)d_out;

    // workspace carve-up (~43.3 MB; fits easily in the 192 MB L2)
    float* ws        = (float*)d_ws;
    float* feat_isr  = ws;                                        // B*160
    float* feat_dec  = feat_isr + (size_t)B_TOTAL * F_ISR;        // B*140
    float* sel       = feat_dec + (size_t)B_TOTAL * F_DEC;        // B*24
    float* partials  = sel + (size_t)B_TOTAL * NSEL * 4;          // 256*2*160
    float* scale_isr = partials + (size_t)STAT_BLOCKS * 2 * F_ISR;
    float* shift_isr = scale_isr + F_ISR;
    float* scale_dec = shift_isr + F_ISR;
    float* shift_dec = scale_dec + F_ISR;
    float* w1p_isr   = shift_dec + F_ISR;                         // 208*160
    float* b1p_isr   = w1p_isr + (size_t)FF_PAD * F_ISR;          // 208
    float* w1p_dec   = b1p_isr + FF_PAD;                          // 208*140
    float* b1p_dec   = w1p_dec + (size_t)FF_PAD * F_DEC;          // 208

    const int rows_per_stat = B_TOTAL / STAT_BLOCKS;              // 128

    // weight padding (independent of everything; launched first)
    pad_w1_kernel<<<(FF_PAD * F_ISR + 255) / 256, 256, 0, stream>>>(
        isr_w1, isr_b1, F_ISR, w1p_isr, b1p_isr);
    pad_w1_kernel<<<(FF_PAD * F_DEC + 255) / 256, 256, 0, stream>>>(
        dec_w1, dec_b1, F_DEC, w1p_dec, b1p_dec);

    // ---------------- ISR head ----------------
    feat_kernel<<<B_TOTAL / 8, 256, 0, stream>>>(
        vectors, NJETS * 4, NJETS, N_ISR, F_ISR,
        isr_cola_w, isr_we, isr_wd, feat_isr);
    stats_kernel<<<STAT_BLOCKS, 256, 0, stream>>>(
        feat_isr, F_ISR, rows_per_stat, partials);
    finalize_kernel<<<1, 256, 0, stream>>>(
        partials, F_ISR, isr_bn_g, isr_bn_b, scale_isr, shift_isr);
    fc_kernel<F_ISR, SA_ISR, NJETS, true><<<B_TOTAL / 16, 128, 0, stream>>>(
        feat_isr, scale_isr, shift_isr, w1p_isr, b1p_isr,
        isr_w2, isr_b2, 0, out, vectors, sel);

    // ---------------- decay head (depends on top-6 selection) ----------------
    feat_kernel<<<B_TOTAL / 8, 256, 0, stream>>>(
        sel, NSEL * 4, NSEL, N_DEC, F_DEC,
        dec_cola_w, dec_we, dec_wd, feat_dec);
    stats_kernel<<<STAT_BLOCKS, 256, 0, stream>>>(
        feat_dec, F_DEC, rows_per_stat, partials);
    finalize_kernel<<<1, 256, 0, stream>>>(
        partials, F_DEC, dec_bn_g, dec_bn_b, scale_dec, shift_dec);
    fc_kernel<F_DEC, SA_DEC, 11, false><<<B_TOTAL / 16, 128, 0, stream>>>(
        feat_dec, scale_dec, shift_dec, w1p_dec, b1p_dec,
        dec_w2, dec_b2, NJETS, out, nullptr, nullptr);
}